// Ive_82952998355348
// MI455X (gfx1250) — compile-verified
//
#include <hip/hip_runtime.h>

// i1e(z) = exp(-|z|) * I1(z), elementwise over fp32.
// Memory-bound streaming kernel for MI455X (gfx1250, wave32):
//   - 2x float4 (global_load_b128 / global_store_b128) per thread
//   - non-temporal hints (1 GB streamed through 192 MB L2, zero reuse)
//   - XLA/Cephes f32 Chebyshev coefficients (matches jax i1e lowering)
//   - v_rcp_f32 / v_rsq_f32 hw approximations (<=1 ulp) for the non-critical
//     divide/rsqrt to keep VALU count under the 23.3 TB/s roofline budget.

typedef float float4v __attribute__((ext_vector_type(4)));

// XLA f32 i1e coefficients (Cephes single precision), small branch (|z| <= 8)
__device__ __forceinline__ float chbevl_A17(float y) {
    const float A[17] = {
        9.38153738649577178388E-9f,  -4.44505912879632808065E-8f,
        2.00329475355213526229E-7f,  -8.56872026469545474066E-7f,
        3.47025130813767847674E-6f,  -1.32731636560394358279E-5f,
        4.78156510755005422638E-5f,  -1.61760815825896745588E-4f,
        5.12285956168575772895E-4f,  -1.51357245063125314899E-3f,
        4.15642294431288815669E-3f,  -1.05640848946261981558E-2f,
        2.47264490306265168283E-2f,  -5.29459812080949914269E-2f,
        1.02643658689847095384E-1f,  -1.76416518357834055153E-1f,
        2.52587186443633654823E-1f};
    float b0 = A[0], b1 = 0.0f, b2 = 0.0f;
#pragma unroll
    for (int i = 1; i < 17; ++i) {
        b2 = b1;
        b1 = b0;
        b0 = __builtin_fmaf(y, b1, A[i] - b2);
    }
    return 0.5f * (b0 - b2);
}

// Large branch (|z| > 8)
__device__ __forceinline__ float chbevl_B7(float y) {
    const float B[7] = {
        -3.83538038596423702205E-9f, -2.63146884688951950684E-8f,
        -2.51223623787020892529E-7f, -3.88256480887769039346E-6f,
        -1.10588938762623716291E-4f, -9.76109749136146840777E-3f,
        7.78576235018280120474E-1f};
    float b0 = B[0], b1 = 0.0f, b2 = 0.0f;
#pragma unroll
    for (int i = 1; i < 7; ++i) {
        b2 = b1;
        b1 = b0;
        b0 = __builtin_fmaf(y, b1, B[i] - b2);
    }
    return 0.5f * (b0 - b2);
}

__device__ __forceinline__ float i1e_f32(float x) {
    float ax = __builtin_fabsf(x);
    // small branch: chbevl(ax/2 - 2, A) * ax
    float ys    = __builtin_fmaf(0.5f, ax, -2.0f);
    float small = chbevl_A17(ys) * ax;
    // large branch: chbevl(32/ax - 2, B) * rsqrt(ax)
    // v_rcp_f32 / v_rsq_f32: <=1 ulp, perturbs result ~1e-7 rel — fine.
    float yl    = __builtin_fmaf(32.0f, __builtin_amdgcn_rcpf(ax), -2.0f);
    float large = chbevl_B7(yl) * __builtin_amdgcn_rsqf(ax);
    float r = (ax <= 8.0f) ? small : large;
    return __builtin_copysignf(r, x);
}

__device__ __forceinline__ float4v i1e_f32x4(float4v v) {
    float4v r;
    r.x = i1e_f32(v.x);
    r.y = i1e_f32(v.y);
    r.z = i1e_f32(v.z);
    r.w = i1e_f32(v.w);
    return r;
}

constexpr int TPB = 256;                      // 8 wave32s per workgroup
constexpr int PER_BLOCK = TPB * 4 * 2;        // 2048 elements per block

__global__ void __launch_bounds__(TPB)
i1e_vec4x2_kernel(const float* __restrict__ in, float* __restrict__ out,
                  int n) {
    const int tile = blockIdx.x * PER_BLOCK;
    const int e0   = tile + (int)threadIdx.x * 4;   // first float4 (elements)
    const int e1   = e0 + TPB * 4;                  // second float4

    if (tile + PER_BLOCK <= n) {
        const float4v* __restrict__ vin = (const float4v*)in;
        float4v*       __restrict__ vout = (float4v*)out;
        // Issue both B128 loads before the FMA body: 8 independent
        // Clenshaw chains per thread, load/compute overlap.
        float4v a = __builtin_nontemporal_load(vin + (e0 >> 2));
        float4v b = __builtin_nontemporal_load(vin + (e1 >> 2));
        float4v ra = i1e_f32x4(a);
        float4v rb = i1e_f32x4(b);
        __builtin_nontemporal_store(ra, vout + (e0 >> 2));
        __builtin_nontemporal_store(rb, vout + (e1 >> 2));
    } else {
        // Tail block: scalar, bounds-checked per element.
        int lim0 = (e0 + 4 < n) ? e0 + 4 : n;
        for (int j = e0; j < lim0; ++j) out[j] = i1e_f32(in[j]);
        int lim1 = (e1 + 4 < n) ? e1 + 4 : n;
        for (int j = e1; j < lim1; ++j) out[j] = i1e_f32(in[j]);
    }
}

extern "C" void kernel_launch(void* const* d_in, const int* in_sizes, int n_in,
                              void* d_out, int out_size, void* d_ws,
                              size_t ws_size, hipStream_t stream) {
    (void)in_sizes; (void)n_in; (void)d_ws; (void)ws_size;
    const float* z = (const float*)d_in[0];
    float* out     = (float*)d_out;
    int n          = out_size;

    int nblocks = (n + PER_BLOCK - 1) / PER_BLOCK;   // 65536 for 134.2M elems
    i1e_vec4x2_kernel<<<nblocks, TPB, 0, stream>>>(z, out, n);
}